// ChamferLoss_12584254177841
// MI455X (gfx1250) — compile-verified
//
#include <hip/hip_runtime.h>

typedef __attribute__((ext_vector_type(2))) float v2f;
typedef __attribute__((ext_vector_type(8))) float v8f;

#define BATCH        4
#define NPTS         8192
#define MPTS         8192
#define WAVES_PER_WG 8
#define WG_THREADS   256
#define FLT_BIG      3.4028235e38f

// ws accumulators: [0] sum of per-pred-row mins, [1] sum of per-ref-col mins,
//                  [2] sum |sdf| terms, [3] sum |color| terms
__global__ void chamfer_zero_acc(float* acc) {
    if (threadIdx.x < 4) acc[threadIdx.x] = 0.0f;
}

// ---------------------------------------------------------------------------
// Pass A: per predicted point, min over all ref points of d2 (chamfer dir 1).
// WMMA computes D' = -2 p.r + |r|^2 (C = inline 0); |p|^2 is re-added per row
// after the min since it is constant along the reduction axis.
//
// LDS holds (-2rx,-2ry,-2rz,|r|^2) per ref point; the B fragment for lane
// half h is exactly the h-th 8-byte half of that float4, so each lane does a
// single ds_load_b64 per tile — no component selects in the hot loop.
// ---------------------------------------------------------------------------
__global__ __launch_bounds__(WG_THREADS, 2)
void chamfer_rowmin(const float* __restrict__ pred_pts,
                    const float* __restrict__ ref_pts,
                    float* __restrict__ acc) {
    extern __shared__ float4 ldsRef[];   // [MPTS] = (-2rx,-2ry,-2rz,|r|^2)

    const int wg_per_batch = (NPTS / 16) / WAVES_PER_WG;   // 64
    const int b    = blockIdx.x / wg_per_batch;
    const int tg   = blockIdx.x % wg_per_batch;
    const int wave = threadIdx.x >> 5;
    const int lane = threadIdx.x & 31;
    const int half = lane >> 4;
    const int l16  = lane & 15;

    // Stage the whole ref cloud for this batch into LDS (128 KB).
    for (int i = threadIdx.x; i < MPTS; i += WG_THREADS) {
        const float* rp = ref_pts + ((size_t)b * MPTS + i) * 3;
        float x = rp[0], y = rp[1], z = rp[2];
        ldsRef[i] = make_float4(-2.0f * x, -2.0f * y, -2.0f * z,
                                x * x + y * y + z * z);
    }
    __syncthreads();

    // Loop-invariant A fragment: 16 predicted rows, row vector (px,py,pz,1).
    // A 16x4 f32 layout: VGPR0 = {K0 | K2}, VGPR1 = {K1 | K3}, row = lane&15.
    const int n_base = (tg * WAVES_PER_WG + wave) * 16;
    const int row    = n_base + l16;
    const float* pp  = pred_pts + ((size_t)b * NPTS + row) * 3;
    float px = pp[0], py = pp[1], pz = pp[2];
    float pq = px * px + py * py + pz * pz;
    v2f a;
    a.x = half ? pz   : px;   // K2 : K0
    a.y = half ? 1.0f : py;   // K3 : K1

    v8f czero;
#pragma unroll
    for (int k = 0; k < 8; ++k) czero[k] = 0.0f;

    v8f rowmin;
#pragma unroll
    for (int k = 0; k < 8; ++k) rowmin[k] = FLT_BIG;

    const v2f* lds2    = (const v2f*)ldsRef;
    const int fragbase = l16 * 2 + half;   // v2f units; loop-invariant

    for (int mt = 0; mt < MPTS / 16; ++mt) {
        v2f bf = lds2[mt * 32 + fragbase];     // one ds_load_b64 per tile
        v8f d = __builtin_amdgcn_wmma_f32_16x16x4_f32(
            false, a, false, bf, (short)0, czero, false, false);
#pragma unroll
        for (int k = 0; k < 8; ++k) rowmin[k] = fminf(rowmin[k], d[k]);
    }

    // Butterfly min over the 16 lanes of each half: afterwards EVERY lane of
    // half h holds the full min for rows n_base + 8h + k in rowmin[k].
#pragma unroll
    for (int k = 0; k < 8; ++k) {
        float v = rowmin[k];
#pragma unroll
        for (int off = 1; off < 16; off <<= 1)
            v = fminf(v, __shfl_xor(v, off, 16));
        rowmin[k] = v;
    }
    // Lane whose own row (l16) lives in this half re-attaches its |p|^2.
    float v = rowmin[0];
#pragma unroll
    for (int k = 1; k < 8; ++k) v = ((l16 & 7) == k) ? rowmin[k] : v;
    bool mine = (l16 >> 3) == half;
    float t = mine ? fmaxf(pq + v, 0.0f) : 0.0f;
#pragma unroll
    for (int off = 1; off < 32; off <<= 1) t += __shfl_xor(t, off, 32);
    if (lane == 0) atomicAdd(&acc[0], t);
}

// ---------------------------------------------------------------------------
// Pass B: per ref point, min+argmin over all predicted points (chamfer dir 2,
// plus the sdf / color L1 gather terms driven by the argmin index).
// WMMA computes D' = |p|^2 - 2 p.r (C = inline 0); |r|^2 is column-constant
// so it is added after the min (argmin is shift-invariant).
// A fragments stream from LDS as direct ds_load_b64 halves of (px,py,pz,|p|^2).
// ---------------------------------------------------------------------------
__global__ __launch_bounds__(WG_THREADS, 2)
void chamfer_colmin(const float* __restrict__ pred_pts,
                    const float* __restrict__ pred_sdfs,
                    const float* __restrict__ pred_cols,
                    const float* __restrict__ ref_pts,
                    const float* __restrict__ ref_sdfs,
                    const float* __restrict__ ref_cols,
                    float* __restrict__ acc) {
    extern __shared__ float4 ldsPred[];  // [NPTS] = (px,py,pz,|p|^2)

    const int wg_per_batch = (MPTS / 16) / WAVES_PER_WG;   // 64
    const int b    = blockIdx.x / wg_per_batch;
    const int tg   = blockIdx.x % wg_per_batch;
    const int wave = threadIdx.x >> 5;
    const int lane = threadIdx.x & 31;
    const int half = lane >> 4;
    const int l16  = lane & 15;

    for (int i = threadIdx.x; i < NPTS; i += WG_THREADS) {
        const float* pp = pred_pts + ((size_t)b * NPTS + i) * 3;
        float x = pp[0], y = pp[1], z = pp[2];
        ldsPred[i] = make_float4(x, y, z, x * x + y * y + z * z);
    }
    __syncthreads();

    // Loop-invariant B fragment: 16 ref columns, col vector (-2rx,-2ry,-2rz,1).
    const int m_base = (tg * WAVES_PER_WG + wave) * 16;
    const int mcol   = m_base + l16;
    const float* rp  = ref_pts + ((size_t)b * MPTS + mcol) * 3;
    float rx = rp[0], ry = rp[1], rz = rp[2];
    float rr = rx * rx + ry * ry + rz * rz;
    v2f bf;
    bf.x = half ? (-2.0f * rz) : (-2.0f * rx);   // K2 : K0
    bf.y = half ? 1.0f         : (-2.0f * ry);   // K3 : K1

    v8f czero;
#pragma unroll
    for (int k = 0; k < 8; ++k) czero[k] = 0.0f;

    float curmin = FLT_BIG;
    int   curidx = 0;

    const v2f* lds2    = (const v2f*)ldsPred;
    const int fragbase = l16 * 2 + half;   // v2f units; loop-invariant

    for (int nt = 0; nt < NPTS / 16; ++nt) {
        v2f a = lds2[nt * 32 + fragbase];      // one ds_load_b64 per tile
        v8f d = __builtin_amdgcn_wmma_f32_16x16x4_f32(
            false, a, false, bf, (short)0, czero, false, false);
        const int rbase = nt * 16 + half * 8;  // D: lanes<16 rows k, else k+8
#pragma unroll
        for (int k = 0; k < 8; ++k) {
            float dv = d[k];
            bool lt  = dv < curmin;
            curidx   = lt ? (rbase + k) : curidx;
            curmin   = lt ? dv : curmin;
        }
    }

    // Combine the two halves (same column lives in lanes l and l+16);
    // prefer the lower index on exact ties (argmin = first occurrence).
    float om = __shfl_xor(curmin, 16, 32);
    int   oi = __shfl_xor(curidx, 16, 32);
    bool take = (om < curmin) || ((om == curmin) && (oi < curidx));
    curmin = take ? om : curmin;
    curidx = take ? oi : curidx;

    if (half == 0) {
        const size_t rb = (size_t)b * MPTS;
        const size_t nb = (size_t)b * NPTS;
        float cmin = fmaxf(curmin + rr, 0.0f);   // re-attach |r|^2, clamp
        float gsdf = ref_sdfs[rb + curidx];
        float sdft = fabsf(gsdf - pred_sdfs[nb + mcol]);
        const float* gc = ref_cols  + (rb + (size_t)curidx) * 3;
        const float* pc = pred_cols + (nb + (size_t)mcol)   * 3;
        float colt = fabsf(gc[0] - pc[0]) + fabsf(gc[1] - pc[1]) +
                     fabsf(gc[2] - pc[2]);
#pragma unroll
        for (int off = 1; off < 16; off <<= 1) {
            cmin += __shfl_xor(cmin, off, 16);
            sdft += __shfl_xor(sdft, off, 16);
            colt += __shfl_xor(colt, off, 16);
        }
        if (l16 == 0) {
            atomicAdd(&acc[1], cmin);
            atomicAdd(&acc[2], sdft);
            atomicAdd(&acc[3], colt);
        }
    }
}

__global__ void chamfer_finalize(const float* __restrict__ acc,
                                 float* __restrict__ out) {
    if (threadIdx.x == 0 && blockIdx.x == 0) {
        float cham = acc[0] * (1.0f / (float)(BATCH * NPTS)) +
                     acc[1] * (1.0f / (float)(BATCH * MPTS));
        float sdf  = acc[2] * (1.0f / (float)(BATCH * MPTS));
        float col  = acc[3] * (1.0f / (float)(BATCH * MPTS * 3));
        out[0] = sdf + col + cham;   // all weights == 1
    }
}

extern "C" void kernel_launch(void* const* d_in, const int* in_sizes, int n_in,
                              void* d_out, int out_size, void* d_ws, size_t ws_size,
                              hipStream_t stream) {
    (void)in_sizes; (void)n_in; (void)out_size; (void)ws_size;
    const float* pred_pts  = (const float*)d_in[0];
    const float* pred_sdfs = (const float*)d_in[1];
    const float* pred_cols = (const float*)d_in[2];
    const float* ref_pts   = (const float*)d_in[3];
    const float* ref_sdfs  = (const float*)d_in[4];
    const float* ref_cols  = (const float*)d_in[5];
    float* acc = (float*)d_ws;
    float* out = (float*)d_out;

    chamfer_zero_acc<<<1, 32, 0, stream>>>(acc);

    const int wgA = BATCH * (NPTS / 16) / WAVES_PER_WG;   // 256 workgroups
    chamfer_rowmin<<<wgA, WG_THREADS, MPTS * sizeof(float4), stream>>>(
        pred_pts, ref_pts, acc);

    const int wgB = BATCH * (MPTS / 16) / WAVES_PER_WG;   // 256 workgroups
    chamfer_colmin<<<wgB, WG_THREADS, NPTS * sizeof(float4), stream>>>(
        pred_pts, pred_sdfs, pred_cols, ref_pts, ref_sdfs, ref_cols, acc);

    chamfer_finalize<<<1, 1, 0, stream>>>(acc, out);
}